// BasisFunction1D_2293512536822
// MI455X (gfx1250) — compile-verified
//
#include <hip/hip_runtime.h>
#include <stdint.h>

// Problem constants (fixed by the reference).
#define I_DIM 128      // INPUT_DIM
#define B_DIM 8192     // BATCH
#define O_DIM 128      // OUTPUT_DIM
#define G_DIM 128      // NUM_GRIDS
#define GP1   129      // NUM_GRIDS + 1

#define BT      16     // batch tile per workgroup
#define THREADS 256    // 8 wave32s, one 16-wide o-tile per wave
#define ROWF    132    // padded row stride in floats (128 + 4), 16B aligned
#define BLKF    268    // padded (g,g+1) block stride in floats (2*132 + 4)
#define NBUF    3      // async pipeline depth (prefetch distance 2)

#define GBUF_FLOATS (NBUF * BT * BLKF)            // 12864
#define GBUF_BYTES  (GBUF_FLOATS * 4)             // 51456
#define PREW_BYTES  (I_DIM * BT * 8)              // 16384 (float2)
#define PREOFF_BYTES (I_DIM * BT * 4)             // 8192  (uint32)
#define SMEM_BYTES  (GBUF_BYTES + PREW_BYTES + PREOFF_BYTES)  // 76032

typedef __attribute__((ext_vector_type(2))) float v2f;
typedef __attribute__((ext_vector_type(8))) float v8f;

// ---------------------------------------------------------------------------
// Gather-interpolate-accumulate.
//   out[o, b] = sum_i (1-d)*Pt[i, idx, o] + d*Pt[i, idx+1, o]
// Each WG: 16 batch elems x all 128 outputs. 3-deep pipelined async DMA
// (GLOBAL_LOAD_ASYNC_TO_LDS_B128 / ASYNCcnt) gathers 1KB (idx,idx+1) row-pair
// blocks into LDS; the MAC work runs on the matrix pipe as
// v_wmma_f32_16x16x4_f32 (co-executes with VALU address generation), with the
// two-hot interpolation weights placed into the B operand per K-chunk.
// ---------------------------------------------------------------------------
__global__ __launch_bounds__(THREADS) void basis_gather_kernel(
    const float* __restrict__ X,        // [I, B]
    const float* __restrict__ Pt,       // [I, G+1, O] (workspace)
    const float* __restrict__ borders,  // [G+1]
    const float* __restrict__ invlen,   // [G]
    float* __restrict__ Out)            // [O, B]
{
  extern __shared__ __align__(16) char smem[];
  float*    gbuf    = (float*)smem;                              // NBUF bufs
  float2*   pre_w   = (float2*)(smem + GBUF_BYTES);              // (1-d, d)
  uint32_t* pre_off = (uint32_t*)(smem + GBUF_BYTES + PREW_BYTES);

  const int t     = threadIdx.x;
  const int b0    = blockIdx.x * BT;
  const int bl    = t & (BT - 1);   // batch lane for the gather phase
  const int p     = t >> 4;         // gather partition 0..15 (64B each of 1KB)
  const int wv    = t >> 5;         // wave id = o-tile (16 outputs)
  const int lane  = t & 31;
  const int n     = lane & 15;      // WMMA column (batch) / A-row m
  const int khalf = lane >> 4;      // 0: fragment K=0,1   1: fragment K=2,3

  // ---- prologue: precompute row offsets + interpolation weights for tile ----
  for (int e = t; e < I_DIM * BT; e += THREADS) {
    const int   i   = e >> 4;
    const int   bb  = e & (BT - 1);
    const float x   = X[(size_t)i * B_DIM + b0 + bb];
    const float ena = __expf(-fabsf(x));
    const float cdf = (x > 0.0f) ? (1.0f - 0.5f * ena) : (0.5f * ena);
    int idx = (int)(cdf * (float)G_DIM);
    idx = idx < 0 ? 0 : (idx > G_DIM - 1 ? G_DIM - 1 : idx);
    const float d = (x - borders[idx]) * invlen[idx];
    pre_off[e] = (uint32_t)((i * GP1 + idx) * (O_DIM * 4));
    pre_w[e]   = make_float2(1.0f - d, d);
  }
  __syncthreads();

  const uint64_t ptbase = (uint64_t)(uintptr_t)Pt;  // uniform -> SGPR pair

  // 4 async B128 gathers = this thread's 64B slice of the 1KB block for step i.
  // The imm offset is added to BOTH the LDS and global address (ISA 08 §4.4);
  // the padded LDS layout mirrors global within each 512B row.
  auto issue_gather = [&](int i, int buf) {
    const uint32_t rowoff = pre_off[i * BT + bl];
    const uint32_t goff   = rowoff + (uint32_t)(p * 64);
    float* ldsp = &gbuf[buf * (BT * BLKF) + bl * BLKF + p * 16 + (p >= 8 ? 4 : 0)];
    const uint32_t ldsa = (uint32_t)(uintptr_t)ldsp;  // low 32 bits = LDS offset
    asm volatile(
        "global_load_async_to_lds_b128 %0, %1, %2 offset:0\n\t"
        "global_load_async_to_lds_b128 %0, %1, %2 offset:16\n\t"
        "global_load_async_to_lds_b128 %0, %1, %2 offset:32\n\t"
        "global_load_async_to_lds_b128 %0, %1, %2 offset:48"
        :: "v"(ldsa), "v"(goff), "s"(ptbase)
        : "memory");
  };

  issue_gather(0, 0);
  issue_gather(1, 1);

  v8f acc = {};   // D/C: 16o x 16b tile, lane n holds 8 rows (m = r or r+8)

  int cur = 0;  // i % NBUF, maintained incrementally
  for (int i = 0; i < I_DIM; ++i) {
    if (i < I_DIM - 1) {
      // outstanding: gather(i), gather(i+1) = 8 instrs; retire gather(i).
      asm volatile("s_wait_asynccnt 4" ::: "memory");
    } else {
      asm volatile("s_wait_asynccnt 0" ::: "memory");
    }
    __syncthreads();  // gather(i) visible; accumulate(i-1) done by all waves

    if (i + 2 < I_DIM) {
      int nxt = cur + 2; if (nxt >= NBUF) nxt -= NBUF;
      issue_gather(i + 2, nxt);  // overwrites buf (i-1)%NBUF: safe after barrier
    }

    const float2 wgt   = pre_w[i * BT + n];
    const float* bbase = &gbuf[cur * (BT * BLKF) + wv * 16 + n];
#pragma unroll
    for (int c = 0; c < 8; ++c) {
      // A: rows (b' = 2c+khalf, side L/R) of the gathered blocks, element o.
      const float* rp = bbase + (2 * c + khalf) * BLKF;
      v2f a; a.x = rp[0]; a.y = rp[ROWF];
      // B: two-hot weights — column n gets (wl, wr) only from its own block.
      const bool sel = (n == 2 * c + khalf);
      v2f wf; wf.x = sel ? wgt.x : 0.0f; wf.y = sel ? wgt.y : 0.0f;
      acc = __builtin_amdgcn_wmma_f32_16x16x4_f32(
          /*neg_a=*/false, a, /*neg_b=*/false, wf,
          /*c_mod=*/(short)0, acc, /*reuse_a=*/false, /*reuse_b=*/false);
    }

    if (++cur >= NBUF) cur = 0;
  }

  // D layout: VGPR r -> M = r (lanes 0-15) or r+8 (lanes 16-31), N = n.
  const int orow0 = wv * 16 + (khalf ? 8 : 0);
#pragma unroll
  for (int r = 0; r < 8; ++r)
    Out[(size_t)(orow0 + r) * B_DIM + b0 + n] = acc[r];
}

// ---------------------------------------------------------------------------
// Transpose P[g][o][i] -> Pt[i][g][o] so a gathered (idx, idx+1) pair is one
// contiguous 1KB block.
// ---------------------------------------------------------------------------
__global__ __launch_bounds__(256) void transpose_P(const float* __restrict__ P,
                                                   float* __restrict__ Pt) {
  __shared__ float tile[32][33];
  const int g  = blockIdx.z;
  const int o0 = blockIdx.x * 32;
  const int i0 = blockIdx.y * 32;
  const int tx = threadIdx.x;   // 0..31
  const int ty = threadIdx.y;   // 0..7
#pragma unroll
  for (int r = 0; r < 4; ++r) {
    const int o = o0 + ty + 8 * r;
    tile[ty + 8 * r][tx] = P[((size_t)g * O_DIM + o) * I_DIM + (i0 + tx)];
  }
  __syncthreads();
#pragma unroll
  for (int r = 0; r < 4; ++r) {
    const int i = i0 + ty + 8 * r;
    Pt[((size_t)i * GP1 + g) * O_DIM + (o0 + tx)] = tile[tx][ty + 8 * r];
  }
}

// ---------------------------------------------------------------------------
extern "C" void kernel_launch(void* const* d_in, const int* in_sizes, int n_in,
                              void* d_out, int out_size, void* d_ws, size_t ws_size,
                              hipStream_t stream) {
  const float* X       = (const float*)d_in[0];  // [128, 8192]
  const float* P       = (const float*)d_in[1];  // [129, 128, 128]
  const float* borders = (const float*)d_in[2];  // [129]
  const float* invlen  = (const float*)d_in[3];  // [128]
  float*       Out     = (float*)d_out;          // [128, 8192]
  float*       Pt      = (float*)d_ws;           // needs 129*128*128*4 = 8.45 MB

  dim3 tb(32, 8);
  dim3 tg(O_DIM / 32, I_DIM / 32, GP1);
  transpose_P<<<tg, tb, 0, stream>>>(P, Pt);

  basis_gather_kernel<<<B_DIM / BT, THREADS, SMEM_BYTES, stream>>>(
      X, Pt, borders, invlen, Out);
}